// APPNP_69604239999351
// MI455X (gfx1250) — compile-verified
//
#include <hip/hip_runtime.h>
#include <hip/hip_bf16.h>

typedef __attribute__((ext_vector_type(2))) float v2f;
typedef __attribute__((ext_vector_type(8))) float v8f;

#define N_NODES  100000
#define N_EDGES  1600000
#define IN_SIZE  512
#define HID_SIZE 256
#define OUT_SIZE 40
#define K_STEPS  10

#define LDS_STRIDE (HID_SIZE + 2)   // pad to avoid 16-way bank conflicts on GEMM2 A-loads

// ---------------------------------------------------------------------------
// Fused MLP encoder: h = relu(X @ W1 + b1) @ W2 + b2, via V_WMMA_F32_16X16X4_F32.
// Block = 256 threads (8 waves), covers a 32-node stripe.
//   GEMM1: wave w -> (mw = w&1) 16-row tile x (nw = w>>1) 64-col group (4 accums).
//   H1 (relu'd) staged to LDS [32 x 256].
//   GEMM2: waves 0..5 -> 2 M-tiles x 3 N-tiles (cols padded 40->48, guarded).
// ---------------------------------------------------------------------------
__launch_bounds__(256)
__global__ void appnp_mlp_wmma(const float* __restrict__ X,
                               const float* __restrict__ W1,
                               const float* __restrict__ b1,
                               const float* __restrict__ W2,
                               const float* __restrict__ b2,
                               float* __restrict__ h,
                               float* __restrict__ h0)
{
    __shared__ float lds_h1[32 * LDS_STRIDE];

    const int tid  = threadIdx.x;
    const int wave = tid >> 5;
    const int lane = tid & 31;
    const int hlf  = lane >> 4;     // 0: lanes 0-15, 1: lanes 16-31
    const int ln   = lane & 15;
    const int rowBase = blockIdx.x * 32;   // 100000 / 32 = 3125 blocks exactly

    // ---------------- GEMM1 ----------------
    const int mw = wave & 1;
    const int nw = wave >> 1;
    const int m0 = rowBase + mw * 16;
    const int nb = nw * 64;

    v8f acc[4];
#pragma unroll
    for (int j = 0; j < 4; ++j) {
        const float bias = b1[nb + 16 * j + ln];   // column broadcast (C layout: all elems share N)
#pragma unroll
        for (int i = 0; i < 8; ++i) acc[j][i] = bias;
    }

    // A fragment: lane ln holds row m0+ln; v0/v1 = K = 2*hlf, 2*hlf+1
    const float* arow = X + (size_t)(m0 + ln) * IN_SIZE;

    for (int kb = 0; kb < IN_SIZE; kb += 4) {
        v2f a = *(const v2f*)(arow + kb + 2 * hlf);   // 8B aligned (even column)
        const float* brow0 = W1 + (size_t)(kb + 2 * hlf) * HID_SIZE;
        const float* brow1 = brow0 + HID_SIZE;
#pragma unroll
        for (int j = 0; j < 4; ++j) {
            v2f b;
            b.x = brow0[nb + 16 * j + ln];
            b.y = brow1[nb + 16 * j + ln];
            acc[j] = __builtin_amdgcn_wmma_f32_16x16x4_f32(
                false, a, false, b, (short)0, acc[j], false, false);
        }
    }

    // ReLU + stage H1 tile into LDS. C/D layout: elem i -> row (i + 8*hlf), col = tile_n + ln
#pragma unroll
    for (int j = 0; j < 4; ++j)
#pragma unroll
        for (int i = 0; i < 8; ++i) {
            const int r = mw * 16 + i + 8 * hlf;
            const int c = nb + 16 * j + ln;
            lds_h1[r * LDS_STRIDE + c] = fmaxf(acc[j][i], 0.0f);
        }

    __syncthreads();

    // ---------------- GEMM2 ----------------
    if (wave < 6) {
        const int mt  = wave & 1;      // 0..1 -> rows 0-15 / 16-31
        const int nt  = wave >> 1;     // 0..2 -> cols 0-15 / 16-31 / 32-47
        const int m0l = mt * 16;
        const int c   = nt * 16 + ln;
        const bool cin = (c < OUT_SIZE);

        v8f acc2;
        const float bias2 = cin ? b2[c] : 0.0f;
#pragma unroll
        for (int i = 0; i < 8; ++i) acc2[i] = bias2;

        for (int kb = 0; kb < HID_SIZE; kb += 4) {
            v2f a = *(const v2f*)&lds_h1[(m0l + ln) * LDS_STRIDE + kb + 2 * hlf];
            v2f b;
            b.x = cin ? W2[(size_t)(kb + 2 * hlf) * OUT_SIZE + c] : 0.0f;
            b.y = cin ? W2[(size_t)(kb + 2 * hlf + 1) * OUT_SIZE + c] : 0.0f;
            acc2 = __builtin_amdgcn_wmma_f32_16x16x4_f32(
                false, a, false, b, (short)0, acc2, false, false);
        }

        if (cin) {
#pragma unroll
            for (int i = 0; i < 8; ++i) {
                const int g = rowBase + m0l + i + 8 * hlf;
                const float v = acc2[i];
                h [(size_t)g * OUT_SIZE + c] = v;
                h0[(size_t)g * OUT_SIZE + c] = v;
            }
        }
    }
}

// ---------------------------------------------------------------------------
// Propagation support kernels (L2-resident working set: ~61 MB << 192 MB L2)
// ---------------------------------------------------------------------------
__global__ void appnp_zero(float* __restrict__ p, int n)
{
    int i = blockIdx.x * blockDim.x + threadIdx.x;
    if (i < n) p[i] = 0.0f;
}

__global__ void appnp_degree(const int* __restrict__ src, const int* __restrict__ dst,
                             float* __restrict__ deg_out, float* __restrict__ deg_in)
{
    int e = blockIdx.x * blockDim.x + threadIdx.x;
    if (e < N_EDGES) {
        atomicAdd(&deg_out[src[e]], 1.0f);
        atomicAdd(&deg_in [dst[e]], 1.0f);
    }
}

__global__ void appnp_norm(float* __restrict__ deg_out, float* __restrict__ deg_in)
{
    int i = blockIdx.x * blockDim.x + threadIdx.x;
    if (i < N_NODES) {
        deg_out[i] = rsqrtf(fmaxf(deg_out[i], 1.0f));   // D_out^{-1/2}
        deg_in [i] = rsqrtf(fmaxf(deg_in [i], 1.0f));   // D_in^{-1/2}
    }
}

// Each edge handled by 8 lanes, 5 consecutive columns per lane (40 = 8*5).
__global__ void appnp_scatter(const float* __restrict__ h,
                              const int*   __restrict__ src,
                              const int*   __restrict__ dst,
                              const float* __restrict__ norm_src,
                              float*       __restrict__ agg)
{
    int t  = blockIdx.x * blockDim.x + threadIdx.x;
    int e  = t >> 3;
    int l8 = t & 7;
    if (e >= N_EDGES) return;
    const int s = src[e];
    const int d = dst[e];
    const float ns = norm_src[s];
    const float* hr = h   + (size_t)s * OUT_SIZE + l8 * 5;
    float*       ar = agg + (size_t)d * OUT_SIZE + l8 * 5;
#pragma unroll
    for (int i = 0; i < 5; ++i)
        atomicAdd(&ar[i], hr[i] * ns);
}

// h <- 0.9 * (agg * D_in^{-1/2}) + 0.1 * h0 ; re-zero agg for the next step.
__global__ void appnp_combine(float* __restrict__ h,
                              float* __restrict__ agg,
                              const float* __restrict__ h0,
                              const float* __restrict__ norm_dst)
{
    int i = blockIdx.x * blockDim.x + threadIdx.x;
    if (i >= N_NODES * OUT_SIZE) return;
    const int n = i / OUT_SIZE;
    const float a = agg[i];
    agg[i] = 0.0f;
    h[i] = 0.9f * (a * norm_dst[n]) + 0.1f * h0[i];
}

// ---------------------------------------------------------------------------
extern "C" void kernel_launch(void* const* d_in, const int* in_sizes, int n_in,
                              void* d_out, int out_size, void* d_ws, size_t ws_size,
                              hipStream_t stream)
{
    const float* X   = (const float*)d_in[0];
    const int*   src = (const int*)  d_in[1];
    const int*   dst = (const int*)  d_in[2];
    const float* W1  = (const float*)d_in[3];
    const float* b1  = (const float*)d_in[4];
    const float* W2  = (const float*)d_in[5];
    const float* b2  = (const float*)d_in[6];
    float* h = (float*)d_out;

    // Workspace layout (all offsets 128B-aligned):
    char* ws = (char*)d_ws;
    float* h0   = (float*)(ws);                         // N*40 f32 = 16,000,000 B
    float* agg  = (float*)(ws + 16000128);              // N*40 f32
    float* nsrc = (float*)(ws + 32000256);              // N f32
    float* ndst = (float*)(ws + 32400384);              // N f32

    const int NH = N_NODES * OUT_SIZE;

    // Fresh state every call (graph-replay safe).
    appnp_zero<<<(NH + 255) / 256, 256, 0, stream>>>(agg, NH);
    appnp_zero<<<(N_NODES + 255) / 256, 256, 0, stream>>>(nsrc, N_NODES);
    appnp_zero<<<(N_NODES + 255) / 256, 256, 0, stream>>>(ndst, N_NODES);

    // Degrees -> symmetric norms.
    appnp_degree<<<(N_EDGES + 255) / 256, 256, 0, stream>>>(src, dst, nsrc, ndst);
    appnp_norm<<<(N_NODES + 255) / 256, 256, 0, stream>>>(nsrc, ndst);

    // Fused WMMA MLP: writes h (d_out) and h0 (workspace).
    appnp_mlp_wmma<<<N_NODES / 32, 256, 0, stream>>>(X, W1, b1, W2, b2, h, h0);

    // K=10 APPNP propagation steps, h updated in place.
    const int scatter_threads = N_EDGES * 8;
    for (int k = 0; k < K_STEPS; ++k) {
        appnp_scatter<<<(scatter_threads + 255) / 256, 256, 0, stream>>>(h, src, dst, nsrc, agg);
        appnp_combine<<<(NH + 255) / 256, 256, 0, stream>>>(h, agg, h0, ndst);
    }
}